// GATv2_41128606827066
// MI455X (gfx1250) — compile-verified
//
#include <hip/hip_runtime.h>

#define B_   2
#define N_   1024
#define N2_  512          // j-half processed per workgroup
#define DN_  128
#define DE_  16
#define MID_ 64
#define OUT_ 128
#define BN_  (B_ * N_)    // 2048 rows

typedef float v2f __attribute__((ext_vector_type(2)));
typedef float v8f __attribute__((ext_vector_type(8)));

static __device__ __forceinline__ v8f wmma4(v2f a, v2f b, v8f c) {
    // V_WMMA_F32_16X16X4_F32 : D = A(16x4) * B(4x16) + C(16x16)
    return __builtin_amdgcn_wmma_f32_16x16x4_f32(false, a, false, b, (short)0, c,
                                                 false, false);
}

// ---------------------------------------------------------------------------
// Generic C[M,Nc] = A[M,K] @ W[K,Nc] + bias   (one wave per 16x16 output tile)
// ---------------------------------------------------------------------------
__global__ __launch_bounds__(128) void lin_wmma_kernel(
    const float* __restrict__ A, const float* __restrict__ W,
    const float* __restrict__ bias, float* __restrict__ C,
    int M, int K, int Nc)
{
    const int lane = threadIdx.x & 31;
    const int w    = threadIdx.x >> 5;
    const int half = lane >> 4;
    const int l16  = lane & 15;
    const int ntN  = Nc >> 4;
    const int ntiles = (M >> 4) * ntN;
    const int tile = blockIdx.x * 4 + w;           // wave-uniform
    if (tile >= ntiles) return;
    const int mt   = tile / ntN;
    const int nt   = tile % ntN;
    const int row0 = mt << 4;
    const int n    = (nt << 4) + l16;

    v8f c;
    const float bv = bias[n];
#pragma unroll
    for (int r = 0; r < 8; ++r) c[r] = bv;

    for (int k0 = 0; k0 < K; k0 += 4) {
        v2f a = *(const v2f*)(A + (size_t)(row0 + l16) * K + k0 + 2 * half);
        v2f bb;
        bb.x = W[(size_t)(k0 + 2 * half) * Nc + n];
        bb.y = W[(size_t)(k0 + 2 * half + 1) * Nc + n];
        c = wmma4(a, bb, c);
    }
#pragma unroll
    for (int r = 0; r < 8; ++r)
        C[(size_t)(row0 + r + 8 * half) * Nc + n] = c[r];
}

// pg = g_features @ Wg + bg   (2x128 @ 128x64, trivially small)
__global__ void pg_kernel(const float* __restrict__ g, const float* __restrict__ Wg,
                          const float* __restrict__ bg, float* __restrict__ pg)
{
    const int t = threadIdx.x;            // 128 threads
    const int b = t >> 6, n = t & 63;
    float acc = bg[n];
    for (int k = 0; k < 128; ++k)         // D_GRAPH = 128
        acc += g[b * 128 + k] * Wg[k * MID_ + n];
    pg[b * MID_ + n] = acc;
}

// ---------------------------------------------------------------------------
// Fused GATv2 attention, flash-style over a 512-j half.
// One workgroup per (b, 16-row i-block, j-half) = 256 WGs, 512 threads, 32KB LDS.
//   pass 1 : logits via WMMA pe-GEMM (+ leaky-relu . Wa scoring) -> LDS [j][i]
//   pass 2a: per-row half-max m_h, half-sum s_h; LDS <- exp(l - m_h)
//   pass 2b: P_h = exp @ values via WMMA, all 16 waves (ntile x K-half split),
//            LDS partial combine; P_h, m_h, s_h -> workspace
// ---------------------------------------------------------------------------
__global__ __launch_bounds__(512) void gat_main(
    const float* __restrict__ efeat, const float* __restrict__ adj,
    const float* __restrict__ We,    const float* __restrict__ be,
    const float* __restrict__ Wa,    const float* __restrict__ ba,
    const float* __restrict__ p1,    const float* __restrict__ p2,
    const float* __restrict__ pg,    const float* __restrict__ values,
    float* __restrict__ part, float* __restrict__ msb, float* __restrict__ ssb)
{
    __shared__ float slog[N2_ * 16];   // [j_local][i_local], 32 KB

    const int lane = threadIdx.x & 31;
    const int w    = threadIdx.x >> 5;        // 16 waves
    const int half = lane >> 4;
    const int l16  = lane & 15;
    const int b    = blockIdx.x >> 7;
    const int i0   = ((blockIdx.x >> 1) & 63) << 4;
    const int h    = blockIdx.x & 1;          // j-half
    const int jb   = h * N2_;                 // global j base

    // ------------------ pass 1: logits ------------------
    {
        const int i = i0 + w;                 // wave w owns row i

        // Persistent B fragments of We (16x64) — 32 VGPRs
        v2f wef[4][4];
#pragma unroll
        for (int kt = 0; kt < 4; ++kt)
#pragma unroll
            for (int nt = 0; nt < 4; ++nt) {
                const int k = kt * 4 + 2 * half;
                const int n = nt * 16 + l16;
                wef[kt][nt].x = We[k * MID_ + n];
                wef[kt][nt].y = We[(k + 1) * MID_ + n];
            }
        float cin[4], waf[4];
#pragma unroll
        for (int nt = 0; nt < 4; ++nt) {
            const int n = nt * 16 + l16;
            cin[nt] = be[n] + p2[((size_t)b * N_ + i) * MID_ + n] + pg[b * MID_ + n];
            waf[nt] = Wa[n];
        }
        const float  ba0    = ba[0];
        const float* erow   = efeat + (((size_t)b * N_ + i) * N_ + jb) * DE_;
        const float* adjrow = adj   + ((size_t)b * N_ + i) * N_ + jb;
        const float* p1row  = p1    + ((size_t)b * N_ + jb) * MID_;

        for (int jt = 0; jt < N2_ / 16; ++jt) {
            const int j0 = jt * 16;
            // prefetch 2 j-tiles ahead (speculative; OOB silently dropped)
            __builtin_prefetch(erow + (size_t)(j0 + 32 + l16) * DE_, 0, 3);

            // A fragments (single-use 128MB stream -> non-temporal)
            v2f af[4];
#pragma unroll
            for (int kt = 0; kt < 4; ++kt)
                af[kt] = __builtin_nontemporal_load(
                    (const v2f*)(erow + (size_t)(j0 + l16) * DE_ + kt * 4 + 2 * half));
            float acc[8];
#pragma unroll
            for (int r = 0; r < 8; ++r) acc[r] = 0.f;

#pragma unroll
            for (int nt = 0; nt < 4; ++nt) {
                v8f c;
#pragma unroll
                for (int r = 0; r < 8; ++r) c[r] = cin[nt];
#pragma unroll
                for (int kt = 0; kt < 4; ++kt)
                    c = wmma4(af[kt], wef[kt][nt], c);
                const int n = nt * 16 + l16;
#pragma unroll
                for (int r = 0; r < 8; ++r) {
                    const int j = j0 + r + 8 * half;  // D row -> local source j
                    float pre = c[r] + p1row[(size_t)j * MID_ + n];
                    float lr  = pre >= 0.f ? pre : 0.01f * pre;  // leaky_relu
                    acc[r]   += lr * waf[nt];
                }
            }
#pragma unroll
            for (int r = 0; r < 8; ++r) {
#pragma unroll
                for (int mask = 1; mask <= 8; mask <<= 1)
                    acc[r] += __shfl_xor(acc[r], mask, 32);
            }
            if (l16 == 0) {                    // lanes 0 and 16 write 8 rows each
#pragma unroll
                for (int r = 0; r < 8; ++r) {
                    const int j = j0 + r + 8 * half;
                    const float av = __builtin_nontemporal_load(adjrow + j);
                    slog[j * 16 + w] = acc[r] + ba0 + (av - 1.0f) * 1e9f;
                }
            }
        }
    }
    __syncthreads();

    // ------------ pass 2a: half-softmax stats (wave w owns row i0+w) ------------
    {
        float m = -3.0e38f;
        for (int j = lane; j < N2_; j += 32) m = fmaxf(m, slog[j * 16 + w]);
#pragma unroll
        for (int mask = 16; mask >= 1; mask >>= 1)
            m = fmaxf(m, __shfl_xor(m, mask, 32));
        float s = 0.f;
        for (int j = lane; j < N2_; j += 32) {
            float p = __expf(slog[j * 16 + w] - m);
            s += p;
        }
#pragma unroll
        for (int mask = 16; mask >= 1; mask >>= 1)
            s += __shfl_xor(s, mask, 32);
        for (int j = lane; j < N2_; j += 32)
            slog[j * 16 + w] = __expf(slog[j * 16 + w] - m);
        if (lane == 0) {
            msb[((size_t)h * B_ + b) * N_ + i0 + w] = m;
            ssb[((size_t)h * B_ + b) * N_ + i0 + w] = s;
        }
    }
    __syncthreads();

    // ---- pass 2b: P_h = exp @ values (all 16 waves: nt = w&7, K-half = w>>3) ----
    {
        const int nt = w & 7;
        const int kh = w >> 3;
        const int n  = nt * 16 + l16;
        const float* vbase = values + ((size_t)b * N_ + jb) * OUT_;

        v8f c;
#pragma unroll
        for (int r = 0; r < 8; ++r) c[r] = 0.f;

        const int kbeg = kh * (N2_ / 2), kend = kbeg + N2_ / 2;
        for (int k0 = kbeg; k0 < kend; k0 += 4) {
            const int j = k0 + 2 * half;
            v2f a, bb;
            a.x  = slog[j * 16 + l16];            // exp[i=l16][j]
            a.y  = slog[(j + 1) * 16 + l16];
            bb.x = vbase[(size_t)j * OUT_ + n];
            bb.y = vbase[(size_t)(j + 1) * OUT_ + n];
            c = wmma4(a, bb, c);
        }

        __syncthreads();                // all reads of slog done; reuse as scratch
        if (kh == 1) {
#pragma unroll
            for (int r = 0; r < 8; ++r) slog[nt * 256 + r * 32 + lane] = c[r];
        }
        __syncthreads();
        if (kh == 0) {
#pragma unroll
            for (int r = 0; r < 8; ++r) {
                float o = c[r] + slog[nt * 256 + r * 32 + lane];
                part[(((size_t)h * B_ + b) * N_ + i0 + r + 8 * half) * OUT_ + n] = o;
            }
        }
    }
}

// ---------------------------------------------------------------------------
// Combine the two j-halves:  out = relu((a1*P1 + a2*P2)/(a1*s1 + a2*s2) + skip)
// ---------------------------------------------------------------------------
__global__ __launch_bounds__(256) void combine_kernel(
    const float* __restrict__ part, const float* __restrict__ msb,
    const float* __restrict__ ssb,  const float* __restrict__ skip,
    float* __restrict__ out)
{
    const size_t idx = (size_t)blockIdx.x * 256 + threadIdx.x;  // < BN_*OUT_
    const size_t row = idx >> 7;                                // b*N + i
    const float m1 = msb[row],              s1 = ssb[row];
    const float m2 = msb[(size_t)BN_ + row], s2 = ssb[(size_t)BN_ + row];
    const float m  = fmaxf(m1, m2);
    const float a1 = __expf(m1 - m), a2 = __expf(m2 - m);
    const float denom = a1 * s1 + a2 * s2;
    const float o = (a1 * part[idx] + a2 * part[(size_t)BN_ * OUT_ + idx]) / denom
                    + skip[idx];
    out[idx] = fmaxf(o, 0.f);
}

// ---------------------------------------------------------------------------
extern "C" void kernel_launch(void* const* d_in, const int* in_sizes, int n_in,
                              void* d_out, int out_size, void* d_ws, size_t ws_size,
                              hipStream_t stream)
{
    (void)in_sizes; (void)n_in; (void)out_size; (void)ws_size;

    const float* features = (const float*)d_in[0];
    const float* efeat    = (const float*)d_in[1];
    const float* gfeat    = (const float*)d_in[2];
    const float* adj      = (const float*)d_in[3];
    const float* Wm  = (const float*)d_in[4];
    const float* bm  = (const float*)d_in[5];
    const float* Wsk = (const float*)d_in[6];
    const float* bsk = (const float*)d_in[7];
    const float* W1  = (const float*)d_in[8];
    const float* b1  = (const float*)d_in[9];
    const float* W2  = (const float*)d_in[10];
    const float* b2  = (const float*)d_in[11];
    const float* We  = (const float*)d_in[12];
    const float* be  = (const float*)d_in[13];
    const float* Wg  = (const float*)d_in[14];
    const float* bg  = (const float*)d_in[15];
    const float* Wa  = (const float*)d_in[16];
    const float* ba  = (const float*)d_in[17];
    float* out = (float*)d_out;

    // workspace (floats): values|skip|p1|p2|pg|part(2 halves)|m|s  (~5.3 MB)
    float* ws      = (float*)d_ws;
    float* valuesw = ws;
    float* skipw   = valuesw + (size_t)BN_ * OUT_;
    float* p1w     = skipw   + (size_t)BN_ * OUT_;
    float* p2w     = p1w     + (size_t)BN_ * MID_;
    float* pgw     = p2w     + (size_t)BN_ * MID_;
    float* partw   = pgw     + (size_t)B_ * MID_;
    float* msw     = partw   + (size_t)2 * BN_ * OUT_;
    float* ssw     = msw     + (size_t)2 * BN_;

    // node-feature linears: M = B*N = 2048 rows, K = 128
    lin_wmma_kernel<<<(128 * 8) / 4, 128, 0, stream>>>(features, Wm,  bm,  valuesw,
                                                       2048, DN_, OUT_);
    lin_wmma_kernel<<<(128 * 8) / 4, 128, 0, stream>>>(features, Wsk, bsk, skipw,
                                                       2048, DN_, OUT_);
    lin_wmma_kernel<<<(128 * 4) / 4, 128, 0, stream>>>(features, W1,  b1,  p1w,
                                                       2048, DN_, MID_);
    lin_wmma_kernel<<<(128 * 4) / 4, 128, 0, stream>>>(features, W2,  b2,  p2w,
                                                       2048, DN_, MID_);
    pg_kernel<<<1, 128, 0, stream>>>(gfeat, Wg, bg, pgw);

    gat_main<<<B_ * (N_ / 16) * 2, 512, 0, stream>>>(efeat, adj, We, be, Wa, ba,
                                                     p1w, p2w, pgw, valuesw,
                                                     partw, msw, ssw);

    combine_kernel<<<(BN_ * OUT_) / 256, 256, 0, stream>>>(partw, msw, ssw,
                                                           skipw, out);
}